// LightweightConv1d_28741921145105
// MI455X (gfx1250) — compile-verified
//
#include <hip/hip_runtime.h>
#include <stdint.h>

typedef __attribute__((ext_vector_type(2))) float v2f;
typedef __attribute__((ext_vector_type(8))) float v8f;

#define NB 8
#define NC 1024
#define NT 4096
#define NH 16
#define NK 7

constexpr int WAVES    = 8;     // waves per block, one channel each
constexpr int TILE_T   = 256;   // t per 16(M)x16(N) WMMA tile group
constexpr int SUBTILES = 4;     // tiles per wave per block
constexpr int BLOCK_T  = TILE_T * SUBTILES;  // 1024
constexpr int LDSF     = 288;   // staged floats per wave: t0-8 .. t0+279

__global__ void __launch_bounds__(256)
lconv1d_wmma_kernel(const float* __restrict__ x,
                    const float* __restrict__ w,
                    const float* __restrict__ bias,
                    float* __restrict__ out) {
  __shared__ float lds[WAVES * LDSF];
  __shared__ float wtab[WAVES * 8];   // 7 softmaxed taps + zero slot, per wave

  const int tid  = threadIdx.x;
  const int wave = tid >> 5;
  const int lane = tid & 31;
  const int n    = lane & 15;   // N column of B/C/D
  const int hi   = lane >> 4;   // half-wave selector

  int blk = blockIdx.x;
  const int tchunk = blk & 3;   blk >>= 2;   // 4 t-chunks of 1024
  const int cgrp   = blk & 127; blk >>= 7;   // 128 channel groups of 8
  const int b      = blk;

  const int c = cgrp * WAVES + wave;
  const int h = c & (NH - 1);
  const long base = ((long)b * NC + c) * NT;
  const float* __restrict__ xc = x + base;
  float* __restrict__ oc = out + base;

  // ---- softmax over the 7 taps of this wave's head ----
  float wt[NK];
  float mx = -1e30f;
#pragma unroll
  for (int k = 0; k < NK; ++k) { wt[k] = w[h * NK + k]; mx = fmaxf(mx, wt[k]); }
  float sum = 0.f;
#pragma unroll
  for (int k = 0; k < NK; ++k) { wt[k] = __expf(wt[k] - mx); sum += wt[k]; }
  const float inv = 1.0f / sum;
#pragma unroll
  for (int k = 0; k < NK; ++k) wt[k] *= inv;

  // ---- publish taps to a per-wave LDS table: wtab[0..6]=taps, wtab[7]=0 ----
  {
    float tv = 0.f;
#pragma unroll
    for (int q = 0; q < NK; ++q)
      if (lane == q) tv = wt[q];
    if (lane < 8) wtab[wave * 8 + lane] = tv;
  }

  // ---- banded-Toeplitz B (24x16) as six 4x16 v2f blocks, via LDS gather ----
  // block kk, slot s: lanes 0-15 hold row K=4kk+s, lanes 16-31 row K=4kk+s+2
  v2f bm[6];
#pragma unroll
  for (int kk = 0; kk < 6; ++kk) {
#pragma unroll
    for (int s = 0; s < 2; ++s) {
      const int j = 4 * kk + s + 2 * hi;
      const int d = j - n;
      const unsigned idx = ((unsigned)d < 7u) ? (unsigned)d : 7u;  // 7 -> 0.0f
      bm[kk][s] = wtab[wave * 8 + idx];
    }
  }

  const float bia = bias[c];
  float* __restrict__ wlds = lds + wave * LDSF;
  const int t0blk = tchunk * BLOCK_T;

  for (int st = 0; st < SUBTILES; ++st) {
    const int t0 = t0blk + st * TILE_T;

    // prefetch next sub-tile toward caches (global_prefetch_b8 path)
    if (st + 1 < SUBTILES)
      __builtin_prefetch(xc + t0 + TILE_T + 8 * lane, 0, 1);

    if (t0 >= 8 && t0 <= NT - 280) {
      // ---- fast interior path: async global->LDS copy, 3 instrs per wave ----
      const float* gsrc = xc + (t0 - 8);
      {
        const int i0 = lane * 4;                         // floats 0..127
        unsigned lo = (unsigned)(uintptr_t)(wlds + i0);
        unsigned long long ga = (unsigned long long)(uintptr_t)(gsrc + i0);
        asm volatile("global_load_async_to_lds_b128 %0, %1, off"
                     :: "v"(lo), "v"(ga) : "memory");
      }
      {
        const int i1 = 128 + lane * 4;                   // floats 128..255
        unsigned lo = (unsigned)(uintptr_t)(wlds + i1);
        unsigned long long ga = (unsigned long long)(uintptr_t)(gsrc + i1);
        asm volatile("global_load_async_to_lds_b128 %0, %1, off"
                     :: "v"(lo), "v"(ga) : "memory");
      }
      {
        const int i2 = 256 + lane;                       // floats 256..287
        unsigned lo = (unsigned)(uintptr_t)(wlds + i2);
        unsigned long long ga = (unsigned long long)(uintptr_t)(gsrc + i2);
        asm volatile("global_load_async_to_lds_b32 %0, %1, off"
                     :: "v"(lo), "v"(ga) : "memory");
      }
      // async LDS writes are NOT ordered with DS reads -> explicit fence
      asm volatile("s_wait_asynccnt 0x0" ::: "memory");
    } else {
      // ---- edge path: guarded scalar stage with zero padding ----
#pragma unroll
      for (int i = 0; i < LDSF / 32; ++i) {
        const int idx = i * 32 + lane;
        const int gt  = t0 - 8 + idx;
        float v = 0.f;
        if (gt >= 0 && gt < NT) v = xc[gt];
        wlds[idx] = v;   // per-wave LDS is in-order with the DS reads below
      }
    }

    // ---- D = A(16x24) x B(24x16) via 6 chained f32 WMMAs; C seeded w/ bias ----
    v8f acc;
#pragma unroll
    for (int i = 0; i < 8; ++i) acc[i] = bia;
#pragma unroll
    for (int kk = 0; kk < 6; ++kk) {
      // A[m=n][j] = x(t0 + 16m + j - 3) = wlds[16m + j + 5]
      const int aidx = 16 * n + 4 * kk + 2 * hi + 5;
      v2f a;
      a[0] = wlds[aidx];
      a[1] = wlds[aidx + 1];
      acc = __builtin_amdgcn_wmma_f32_16x16x4_f32(
          /*neg_a=*/false, a, /*neg_b=*/false, bm[kk],
          /*c_mod=*/(short)0, acc, /*reuse_a=*/false, /*reuse_b=*/false);
    }

    // ---- store: VGPR v holds out[t0 + 16v + 128*hi + n] ----
#pragma unroll
    for (int v = 0; v < 8; ++v)
      oc[t0 + 16 * v + 128 * hi + n] = acc[v];
  }
}

extern "C" void kernel_launch(void* const* d_in, const int* in_sizes, int n_in,
                              void* d_out, int out_size, void* d_ws, size_t ws_size,
                              hipStream_t stream) {
  const float* x    = (const float*)d_in[0];   // (8,1024,4096)
  const float* w    = (const float*)d_in[1];   // (16,1,7)
  const float* bias = (const float*)d_in[2];   // (1024,)
  float* out        = (float*)d_out;           // (8,1024,4096)

  const int grid = NB * (NC / WAVES) * (NT / BLOCK_T);  // 8*128*4 = 4096
  lconv1d_wmma_kernel<<<grid, 256, 0, stream>>>(x, w, bias, out);
}